// filter_deep_phase_19988777795974
// MI455X (gfx1250) — compile-verified
//
#include <hip/hip_runtime.h>
#include <math.h>

// ---------------------------------------------------------------------------
// MI455X (gfx1250) implementation of the 13-layer phase-filter CNN.
//
// Design (see analysis): ~100 GFLOP total, working set (<=114 MB ping-pong)
// is resident in the 192 MB L2 -> the pipeline is L2-BW / latency bound.
// Convs are mapped to fp32 WMMA (V_WMMA_F32_16X16X4_F32): per dilated tap,
// a GEMM with M=co (<=16), K=ci (chunks of 4), N=16 output columns per wave.
// fp32 WMMA keeps exact fp32 MAC semantics vs the fp32 reference (13 BN+ELU
// stages compound error; f16 would cut matrix-instr count 8x but is not
// needed at this FLOP scale).
//
// BN is a deterministic two-stage reduction (no float atomics -> identical
// results on every graph replay), finalized in double and fused to a
// per-channel (scale, shift) pair. Conv bias is skipped: BN subtracts the
// channel mean, which cancels any constant per-channel bias exactly.
// ---------------------------------------------------------------------------

typedef __attribute__((ext_vector_type(2))) float v2f;
typedef __attribute__((ext_vector_type(8))) float v8f;

#define HWD   63
#define NBATCH 256
#define ADDC  (-2.1415f)
#define EPSC  (1e-05f)

// ---------------------------------------------------------------------------
// Extract phase channel: x is (256, 2, 63, 63); phase = x[:, 1, :, :]
// ---------------------------------------------------------------------------
__global__ __launch_bounds__(256) void extract_phase(const float* __restrict__ x,
                                                     float* __restrict__ out) {
    int idx = blockIdx.x * 256 + threadIdx.x;
    const int npx = HWD * HWD;
    if (idx < NBATCH * npx) {
        int n = idx / npx, p = idx % npx;
        out[idx] = x[(n * 2 + 1) * npx + p];
    }
}

// ---------------------------------------------------------------------------
// Implicit-GEMM dilated conv via V_WMMA_F32_16X16X4_F32.
//   grid.x = NBATCH * outH  (one block per output row of one image)
//   block  = 128 threads = 4 waves; wave w covers output columns [16w,16w+16)
// Fragment layouts per CDNA5 ISA 7.12.2 (wave32):
//   A (16x4 f32):  lane L -> M = L%16; K pair = {0,1} (L<16) or {2,3}
//   B (4x16 f32):  lane L -> N = L%16; K pair = {0,1} (L<16) or {2,3}
//   C (16x16 f32): VGPR r, lane L -> M = r + (L<16 ? 0 : 8), N = L%16
// Zero-padding of the conv and K/M padding to 4/16 are handled by zeroing
// fragment elements, so EXEC stays all-ones around every WMMA (ISA req).
// ---------------------------------------------------------------------------
__global__ __launch_bounds__(128) void conv_wmma(
    const float* __restrict__ in, const float* __restrict__ wgt,
    float* __restrict__ out,
    int ci, int co, int ks, int pad, int dil,
    int inH, int inW, int outH, int outW) {

    const int bx   = blockIdx.x;
    const int n    = bx / outH;
    const int oh   = bx % outH;
    const int lane = threadIdx.x & 31;
    const int wv   = threadIdx.x >> 5;
    const int nn   = lane & 15;          // N (column) / M (row) sub-index
    const int hi   = lane >> 4;          // lane half
    const int kb   = hi * 2;             // K sub-base inside the 4-chunk
    const int ow   = wv * 16 + nn;       // output column this lane produces
    const int m    = nn;                 // A-matrix row (output channel)

    v8f acc = {0.f, 0.f, 0.f, 0.f, 0.f, 0.f, 0.f, 0.f};

    const float* inN = in + (size_t)n * ci * inH * inW;
    const bool mOK = (m < co);
    const float* wm = wgt + (size_t)m * ci * ks * ks;

    for (int c0 = 0; c0 < ci; c0 += 4) {
        const int  cA0 = c0 + kb, cA1 = c0 + kb + 1;
        const bool v0  = (cA0 < ci), v1 = (cA1 < ci);
        const float* wr0 = wm + (size_t)cA0 * ks * ks;
        const float* wr1 = wm + (size_t)cA1 * ks * ks;

        for (int ky = 0; ky < ks; ++ky) {
            const int  ih  = oh + ky * dil - pad;
            const bool hOK = ((unsigned)ih < (unsigned)inH);
            const float* row0 = inN + ((size_t)cA0 * inH + ih) * inW;
            const float* row1 = inN + ((size_t)cA1 * inH + ih) * inW;
            // prefetch next tap row (emits global_prefetch_b8; speculative, safe)
            __builtin_prefetch(row0 + (size_t)dil * inW, 0, 0);

            for (int kx = 0; kx < ks; ++kx) {
                const int  iw  = ow + kx * dil - pad;
                const bool wOK = ((unsigned)iw < (unsigned)inW);
                const int  t   = ky * ks + kx;

                float a0 = (mOK && v0) ? wr0[t] : 0.f;
                float a1 = (mOK && v1) ? wr1[t] : 0.f;
                float b0 = (hOK && wOK && v0) ? row0[iw] : 0.f;
                float b1 = (hOK && wOK && v1) ? row1[iw] : 0.f;

                v2f A = {a0, a1};
                v2f B = {b0, b1};
                acc = __builtin_amdgcn_wmma_f32_16x16x4_f32(
                    /*neg_a=*/false, A, /*neg_b=*/false, B,
                    /*c_mod=*/(short)0, acc, /*reuse_a=*/false, /*reuse_b=*/false);
            }
        }
    }

    if (ow < outW) {
        const int mBase = hi * 8;
        #pragma unroll
        for (int r = 0; r < 8; ++r) {
            int mm = mBase + r;
            if (mm < co)
                out[(((size_t)n * co + mm) * outH + oh) * outW + ow] = acc[r];
        }
    }
}

// ---------------------------------------------------------------------------
// Locally-connected layer: out[n,h,w] = sum_c in[n,c,h,w] * w[c,h,w]
// (weights vary per pixel -> not GEMM shaped; plain VALU, C <= 16)
// ---------------------------------------------------------------------------
__global__ __launch_bounds__(256) void lc_kernel(const float* __restrict__ in,
                                                 const float* __restrict__ w,
                                                 float* __restrict__ out, int C) {
    const int npx = HWD * HWD;
    int idx = blockIdx.x * 256 + threadIdx.x;
    if (idx < NBATCH * npx) {
        int n = idx / npx, p = idx % npx;
        float s = 0.f;
        for (int c = 0; c < C; ++c)
            s += in[((size_t)n * C + c) * npx + p] * w[(size_t)c * npx + p];
        out[idx] = s;
    }
}

// ---------------------------------------------------------------------------
// BN stage 1: per-(channel, batch-image) partial sum / sum-of-squares.
// grid = (NBATCH, C); deterministic tree reduction in LDS.
// ---------------------------------------------------------------------------
__global__ __launch_bounds__(256) void bn_stats_partial(const float* __restrict__ y,
                                                        float2* __restrict__ part,
                                                        int C, int Npix) {
    const int n = blockIdx.x, c = blockIdx.y, tid = threadIdx.x;
    const float* p = y + ((size_t)n * C + c) * Npix;
    float s = 0.f, ss = 0.f;
    for (int i = tid; i < Npix; i += 256) {
        float v = p[i];
        s += v;
        ss += v * v;
    }
    __shared__ float sh[256], sh2[256];
    sh[tid] = s; sh2[tid] = ss;
    __syncthreads();
    for (int o = 128; o > 0; o >>= 1) {
        if (tid < o) { sh[tid] += sh[tid + o]; sh2[tid] += sh2[tid + o]; }
        __syncthreads();
    }
    if (tid == 0) part[(size_t)c * gridDim.x + n] = make_float2(sh[0], sh2[0]);
}

// ---------------------------------------------------------------------------
// BN stage 2: finalize in double, fuse gamma/beta into (scale, shift).
//   scale = g / sqrt(var + eps), shift = be - mean * scale
// ---------------------------------------------------------------------------
__global__ __launch_bounds__(256) void bn_stats_final(const float2* __restrict__ part,
                                                      const float* __restrict__ g,
                                                      const float* __restrict__ be,
                                                      float2* __restrict__ chp,
                                                      int nblk, int Npix) {
    const int c = blockIdx.x, tid = threadIdx.x;
    double s = 0.0, ss = 0.0;
    for (int i = tid; i < nblk; i += 256) {
        float2 v = part[(size_t)c * nblk + i];
        s += (double)v.x; ss += (double)v.y;
    }
    __shared__ double sh[256], sh2[256];
    sh[tid] = s; sh2[tid] = ss;
    __syncthreads();
    for (int o = 128; o > 0; o >>= 1) {
        if (tid < o) { sh[tid] += sh[tid + o]; sh2[tid] += sh2[tid + o]; }
        __syncthreads();
    }
    if (tid == 0) {
        double cnt  = (double)nblk * (double)Npix;
        double mean = sh[0] / cnt;
        double var  = sh2[0] / cnt - mean * mean;
        double invs = 1.0 / sqrt(var + (double)EPSC);
        float scale = (float)(invs * (double)g[c]);
        float shift = (float)((double)be[c] - mean * invs * (double)g[c]);
        chp[c] = make_float2(scale, shift);
    }
}

// ---------------------------------------------------------------------------
// BN apply + ELU + ADD (in place):  v = y*scale + shift; elu(v) + ADD
// ---------------------------------------------------------------------------
__global__ __launch_bounds__(256) void bn_apply(float* __restrict__ y,
                                                const float2* __restrict__ chp,
                                                int C, int Npix, int total) {
    int idx = blockIdx.x * 256 + threadIdx.x;
    if (idx < total) {
        int c = (idx / Npix) % C;
        float2 p = chp[c];
        float v = y[idx] * p.x + p.y;
        v = (v > 0.f) ? v : (expf(v) - 1.f);
        y[idx] = v + ADDC;
    }
}

// ---------------------------------------------------------------------------
// Final: phase = act + x[:,1]; point-symmetry reflection.
// cond(i,j) = (i>31) | (i==31 & j>31)  ->  -phase[62-i, 62-j], else phase[i,j]
// ---------------------------------------------------------------------------
__global__ __launch_bounds__(256) void final_sym(const float* __restrict__ act,
                                                 const float* __restrict__ x,
                                                 float* __restrict__ out) {
    const int npx = HWD * HWD;
    int idx = blockIdx.x * 256 + threadIdx.x;
    if (idx < NBATCH * npx) {
        int n = idx / npx, p = idx % npx;
        int i = p / HWD, j = p % HWD;
        const int c = HWD / 2;
        bool cond = (i > c) || (i == c && j > c);
        int pr = (HWD - 1 - i) * HWD + (HWD - 1 - j);
        const float* xn = x + ((size_t)n * 2 + 1) * npx;
        const float* an = act + (size_t)n * npx;
        float v = cond ? -(an[pr] + xn[pr]) : (an[p] + xn[p]);
        out[(size_t)n * npx + p] = v;
    }
}

// ---------------------------------------------------------------------------
// Host-side pipeline. All launches on `stream`; graph-capture safe.
// ---------------------------------------------------------------------------
extern "C" void kernel_launch(void* const* d_in, const int* in_sizes, int n_in,
                              void* d_out, int out_size, void* d_ws, size_t ws_size,
                              hipStream_t stream) {
    const float* x = (const float*)d_in[0];
    char* ws = (char*)d_ws;

    // ping-pong activation buffers (bufA sees max 16 ch, bufB max 12 ch)
    const size_t npx63  = (size_t)HWD * HWD;
    const size_t bufAsz = (size_t)16 * NBATCH * npx63 * sizeof(float);
    const size_t bufBsz = (size_t)12 * NBATCH * npx63 * sizeof(float);
    float*  bufA = (float*)ws;
    float*  bufB = (float*)(ws + bufAsz);
    float2* part = (float2*)(ws + bufAsz + bufBsz);
    float2* chp  = (float2*)(ws + bufAsz + bufBsz + (size_t)16 * NBATCH * sizeof(float2));

    // layer table: param indices follow setup_inputs() dict insertion order
    struct L { int isConv, ci, co, ks, pad, dil, wI, gI, beI; };
    static const L Ls[13] = {
        {1,  1,  4, 23, 11, 1,  1,  3,  4},
        {1,  4,  8, 21, 10, 1,  5,  7,  8},
        {1,  8, 12, 17,  8, 1,  9, 11, 12},
        {0, 12,  1,  0,  0, 0, 13, 14, 15},
        {1,  1,  4,  5,  3, 2, 16, 18, 19},
        {1,  4,  8,  5,  2, 1, 20, 22, 23},
        {1,  8, 12,  3,  3, 2, 24, 26, 27},
        {1, 12, 16,  3,  1, 1, 28, 30, 31},
        {0, 16,  1,  0,  0, 0, 32, 33, 34},
        {1,  1,  4,  3,  1, 1, 35, 37, 38},
        {1,  4,  8,  3,  1, 1, 39, 41, 42},
        {1,  8, 12,  3,  2, 2, 43, 45, 46},
        {0, 12,  1,  0,  0, 0, 47, 48, 49},
    };

    float* cur = bufA;
    float* nxt = bufB;
    int H = HWD;

    {
        int tot = NBATCH * (int)npx63;
        extract_phase<<<(tot + 255) / 256, 256, 0, stream>>>(x, cur);
    }

    for (int i = 0; i < 13; ++i) {
        const L& s = Ls[i];
        int C, outH;
        if (s.isConv) {
            outH = H + 2 * s.pad - (s.dil * (s.ks - 1) + 1) + 1;  // 63 or 61
            conv_wmma<<<dim3(NBATCH * outH), 128, 0, stream>>>(
                cur, (const float*)d_in[s.wI], nxt,
                s.ci, s.co, s.ks, s.pad, s.dil, H, H, outH, outH);
            C = s.co;
        } else {
            outH = H;  // always 63 at lc layers
            int tot = NBATCH * (int)npx63;
            lc_kernel<<<(tot + 255) / 256, 256, 0, stream>>>(
                cur, (const float*)d_in[s.wI], nxt, s.ci);
            C = 1;
        }
        int Npix = outH * outH;
        bn_stats_partial<<<dim3(NBATCH, C), 256, 0, stream>>>(nxt, part, C, Npix);
        bn_stats_final<<<dim3(C), 256, 0, stream>>>(
            part, (const float*)d_in[s.gI], (const float*)d_in[s.beI], chp, NBATCH, Npix);
        int tot = NBATCH * C * Npix;
        bn_apply<<<(tot + 255) / 256, 256, 0, stream>>>(nxt, chp, C, Npix, tot);

        float* t = cur; cur = nxt; nxt = t;
        H = outH;
    }

    {
        int tot = NBATCH * (int)npx63;
        final_sym<<<(tot + 255) / 256, 256, 0, stream>>>(cur, x, (float*)d_out);
    }
}